// RGCNHighMemConv_8409545965795
// MI455X (gfx1250) — compile-verified
//
#include <hip/hip_runtime.h>

#define IN_FEAT  64
#define OUT_FEAT 64
#define NUM_RELS 64
#define TILE_M   16
#define LDS_STRIDE 65   // 64 + 1 pad -> conflict-free column access

typedef __attribute__((ext_vector_type(2))) float v2f;
typedef __attribute__((ext_vector_type(8))) float v8f;

// ---------------- Phase 0: init ----------------
__global__ void zero_out_kernel(float* __restrict__ out, int n) {
    int i = blockIdx.x * blockDim.x + threadIdx.x;
    int stride = gridDim.x * blockDim.x;
    for (; i < n; i += stride) out[i] = 0.0f;
}

__global__ void init_counts_kernel(int* __restrict__ counts, int* __restrict__ cursor) {
    int t = threadIdx.x;
    if (t < NUM_RELS) { counts[t] = 0; cursor[t] = 0; }
}

// ---------------- Phase 1: bin edges by relation ----------------
__global__ void count_kernel(const int* __restrict__ etypes, int E, int* __restrict__ counts) {
    int e = blockIdx.x * blockDim.x + threadIdx.x;
    if (e < E) atomicAdd(&counts[etypes[e]], 1);
}

__global__ void scan_kernel(const int* __restrict__ counts,
                            int* __restrict__ offsets,
                            int* __restrict__ tile_offsets) {
    if (threadIdx.x == 0) {
        int acc = 0, tacc = 0;
        for (int r = 0; r < NUM_RELS; ++r) {
            offsets[r]      = acc;
            tile_offsets[r] = tacc;
            acc  += counts[r];
            tacc += (counts[r] + TILE_M - 1) / TILE_M;   // per-relation tile padding
        }
        offsets[NUM_RELS]      = acc;
        tile_offsets[NUM_RELS] = tacc;
    }
}

__global__ void scatter_kernel(const int* __restrict__ etypes, int E,
                               const int* __restrict__ offsets,
                               int* __restrict__ cursor,
                               int* __restrict__ edge_ids) {
    int e = blockIdx.x * blockDim.x + threadIdx.x;
    if (e < E) {
        int r = etypes[e];
        int pos = offsets[r] + atomicAdd(&cursor[r], 1);
        edge_ids[pos] = e;
    }
}

// ---------------- Phase 2: tiled WMMA GEMM + scatter-add ----------------
// Block = 128 threads (4 wave32). One block per 16-edge tile of one relation.
// Wave w computes output columns [16w, 16w+16) of the 16x64 msg tile via
// 16 chained V_WMMA_F32_16X16X4_F32 (K = 64).
__global__ __launch_bounds__(128)
void rgcn_wmma_kernel(const float* __restrict__ feat,
                      const float* __restrict__ weight,
                      const float* __restrict__ norm,
                      const int*   __restrict__ src,
                      const int*   __restrict__ dst,
                      const int*   __restrict__ offsets,
                      const int*   __restrict__ tile_offsets,
                      const int*   __restrict__ edge_ids,
                      float*       __restrict__ out) {
    __shared__ int   s_to[NUM_RELS + 1];
    __shared__ int   s_dst[TILE_M];
    __shared__ int   s_srcrow[TILE_M];
    __shared__ int   s_valid[TILE_M];
    __shared__ float s_norm[TILE_M];
    __shared__ float sA[TILE_M * LDS_STRIDE];   // 16 x 64 feat rows, padded

    const int t = threadIdx.x;
    if (t <= NUM_RELS) s_to[t] = tile_offsets[t];
    __syncthreads();

    const int total_tiles = s_to[NUM_RELS];
    const int b = blockIdx.x;
    if (b >= total_tiles) return;                 // uniform per block

    // find relation r with tile_offsets[r] <= b < tile_offsets[r+1]
    int r = 0;
    while (r + 1 < NUM_RELS && s_to[r + 1] <= b) ++r;

    const int lt    = b - s_to[r];                // local tile within relation
    const int ebase = offsets[r] + lt * TILE_M;
    const int eend  = offsets[r + 1];

    if (t < TILE_M) {
        int ei = ebase + t;
        if (ei < eend) {
            int e       = edge_ids[ei];
            s_valid[t]  = 1;
            s_dst[t]    = dst[e];
            s_norm[t]   = norm[e];
            s_srcrow[t] = src[e];
        } else {
            s_valid[t]  = 0;
            s_dst[t]    = 0;
            s_norm[t]   = 0.0f;
            s_srcrow[t] = -1;                     // -> zero A row, keeps EXEC all-1s
        }
    }
    __syncthreads();

    // Stage gathered feat rows into LDS: 1024 floats, 8 per thread (row-contiguous)
    {
        int flat = t * 8;                         // 0..1023
        int m = flat >> 6;
        int k = flat & 63;
        int srow = s_srcrow[m];
        if (srow >= 0) {
            const float* p = feat + (size_t)srow * IN_FEAT + k;
            #pragma unroll
            for (int j = 0; j < 8; ++j) sA[m * LDS_STRIDE + k + j] = p[j];
        } else {
            #pragma unroll
            for (int j = 0; j < 8; ++j) sA[m * LDS_STRIDE + k + j] = 0.0f;
        }
    }
    __syncthreads();

    const int lane    = t & 31;
    const int wave    = t >> 5;                   // 0..3
    const int nbase   = wave * 16;
    const int halfsel = lane >> 4;                // 0: lanes 0-15, 1: lanes 16-31
    const int lm      = lane & 15;

    v8f c = {0.f, 0.f, 0.f, 0.f, 0.f, 0.f, 0.f, 0.f};

    const float* wr = weight + (size_t)r * IN_FEAT * OUT_FEAT;

    // K = 64 in steps of 4. A-fragment (ISA 16x4 f32 layout):
    //   lane = M (0-15 / 16-31 mirror), VGPR j holds K = kb + j, kb = 4*kk + 2*halfsel
    // B-fragment (row-striped): lane = N, VGPR j holds K = kb + j
    #pragma unroll
    for (int kk = 0; kk < IN_FEAT / 4; ++kk) {
        const int kb = kk * 4 + 2 * halfsel;
        v2f a, bm;
        a.x  = sA[lm * LDS_STRIDE + kb + 0];
        a.y  = sA[lm * LDS_STRIDE + kb + 1];
        bm.x = wr[(kb + 0) * OUT_FEAT + nbase + lm];
        bm.y = wr[(kb + 1) * OUT_FEAT + nbase + lm];
        c = __builtin_amdgcn_wmma_f32_16x16x4_f32(
                /*neg_a=*/false, a, /*neg_b=*/false, bm,
                /*c_mod=*/(short)0, c, /*reuse_a=*/false, /*reuse_b=*/false);
    }

    // Writeback: C/D layout -> lane gives N = nbase+lm, VGPR j gives M = j + 8*halfsel.
    #pragma unroll
    for (int j = 0; j < 8; ++j) {
        int m = j + 8 * halfsel;
        if (s_valid[m]) {
            float v = c[j] * s_norm[m];
            atomicAdd(&out[(size_t)s_dst[m] * OUT_FEAT + nbase + lm], v);
        }
    }
}

// ---------------- Launcher ----------------
extern "C" void kernel_launch(void* const* d_in, const int* in_sizes, int n_in,
                              void* d_out, int out_size, void* d_ws, size_t ws_size,
                              hipStream_t stream) {
    const float* feat   = (const float*)d_in[0];
    const float* weight = (const float*)d_in[1];
    const float* norm   = (const float*)d_in[2];
    const int*   src    = (const int*)d_in[3];
    const int*   dst    = (const int*)d_in[4];
    const int*   etypes = (const int*)d_in[5];
    float*       out    = (float*)d_out;
    const int E = in_sizes[3];

    // Workspace layout (ints): counts[64] | offsets[65] | tile_offsets[65] | cursor[64] | edge_ids[E]
    int* ws           = (int*)d_ws;
    int* counts       = ws;
    int* offsets      = ws + 64;
    int* tile_offsets = ws + 192;
    int* cursor       = ws + 320;
    int* edge_ids     = ws + 384;
    (void)ws_size; (void)n_in;

    zero_out_kernel<<<512, 256, 0, stream>>>(out, out_size);
    init_counts_kernel<<<1, 64, 0, stream>>>(counts, cursor);
    count_kernel<<<(E + 255) / 256, 256, 0, stream>>>(etypes, E, counts);
    scan_kernel<<<1, 32, 0, stream>>>(counts, offsets, tile_offsets);
    scatter_kernel<<<(E + 255) / 256, 256, 0, stream>>>(etypes, E, offsets, cursor, edge_ids);

    // Upper bound on padded tile count: sum ceil(c_r/16) <= E/16 + NUM_RELS
    const int max_tiles = (E + TILE_M - 1) / TILE_M + NUM_RELS;
    rgcn_wmma_kernel<<<max_tiles, 128, 0, stream>>>(feat, weight, norm, src, dst,
                                                    offsets, tile_offsets, edge_ids, out);
}